// GATv2WithGlobal_3143916060992
// MI455X (gfx1250) — compile-verified
//
#include <hip/hip_runtime.h>
#include <math.h>

// ---------------- problem constants (from reference) ----------------
#define N_NODES 50000
#define N_EDGES 800000
#define NB      512
#define F_IN    32
#define G_IN    16
#define CC      32      // channels per head
#define NH      4       // heads
#define HC      128     // NH*CC
#define ET      (N_EDGES + N_NODES)   // edges incl. self loops
#define NEG_SLOPE 0.2f
#define BN_EPS    1e-5f

typedef float v2f __attribute__((ext_vector_type(2)));
typedef float v8f __attribute__((ext_vector_type(8)));

// monotone float <-> u32 mapping so segment-max can use integer atomicMax
__device__ __forceinline__ unsigned ordF(float f) {
  unsigned u = __float_as_uint(f);
  return (u & 0x80000000u) ? ~u : (u | 0x80000000u);
}
__device__ __forceinline__ float unordF(unsigned u) {
  unsigned v = (u & 0x80000000u) ? (u & 0x7fffffffu) : ~u;
  return __uint_as_float(v);
}

// ---------------- generic fill ----------------
__global__ void k_fill(unsigned* __restrict__ p, unsigned val, int n) {
  int i = blockIdx.x * 256 + threadIdx.x;
  if (i < n) p[i] = val;
}

// ---------------- fp32 WMMA GEMM: Y[N,128] = X[N,DIN] @ W[DIN,128] + bias ----
// Block = 256 threads = 8 waves; block covers 16 rows x 128 cols (wave = 16x16 tile).
// N_NODES is an exact multiple of 16 and cols are exactly 8 tiles, so the
// epilogue is branch-free (no exec-mask churn, single bias load per lane).
// Uses V_WMMA_F32_16X16X4_F32. VGPR layouts per CDNA5 ISA 7.12.2:
//   A 16x4 f32 : v0 = K=0 (lanes 0-15) / K=2 (lanes 16-31), v1 = K=1 / K=3
//   B 4x16 f32 : mirrored (N = lane%16, K halves split across lane halves)
//   D 16x16    : VGPR r -> row r (lanes 0-15) / row r+8 (lanes 16-31)
template <int DIN>
__global__ void k_gemm_wmma(const float* __restrict__ X, const float* __restrict__ W,
                            const float* __restrict__ bias, float* __restrict__ Y) {
  __shared__ float sx[16 * (DIN + 4)];   // +4 pad: spreads LDS banks for f2 loads
  const int row0 = blockIdx.x * 16;
  const int tid  = threadIdx.x;

#pragma unroll
  for (int i = 0; i < 16 * DIN / 256; ++i) {
    int idx = i * 256 + tid;
    int r = idx / DIN, c = idx - r * DIN;
    sx[r * (DIN + 4) + c] = X[(size_t)(row0 + r) * DIN + c];
  }
  __syncthreads();

  const int wid  = tid >> 5;
  const int lane = tid & 31;
  const int half = lane >> 4;       // 0 or 1
  const int nn   = lane & 15;       // row (for A) / col (for B/D) within tile
  const int col0 = wid * 16;
  const int col  = col0 + nn;

  v8f d = {0.f, 0.f, 0.f, 0.f, 0.f, 0.f, 0.f, 0.f};
#pragma unroll
  for (int k0 = 0; k0 < DIN; k0 += 4) {
    const int ka = k0 + 2 * half;
    v2f a, b;
    a.x = sx[nn * (DIN + 4) + ka];
    a.y = sx[nn * (DIN + 4) + ka + 1];
    b.x = W[ka * HC + col];
    b.y = W[(ka + 1) * HC + col];
    d = __builtin_amdgcn_wmma_f32_16x16x4_f32(false, a, false, b, (short)0, d,
                                              false, false);
  }

  const float bv = bias[col];
  float* yb = Y + (size_t)(row0 + 8 * half) * HC + col;
#pragma unroll
  for (int r = 0; r < 8; ++r) {
    yb[(size_t)r * HC] = d[r] + bv;
  }
}

// ---------------- edge pass 1: logits + per-target max --------------
// wave per edge; lane handles 4 channels (float4). head = lane>>3 (8 lanes/head).
__global__ void k_edge_logit(const float4* __restrict__ xl, const float4* __restrict__ xr,
                             const int* __restrict__ esrc, const int* __restrict__ edst,
                             const float4* __restrict__ att4,
                             float* __restrict__ logit, unsigned* __restrict__ mord) {
  const int wid = threadIdx.x >> 5, lane = threadIdx.x & 31;
  const int e = blockIdx.x * 8 + wid;
  if (e >= ET) return;
  int src, dst;
  if (e < N_EDGES) { src = esrc[e]; dst = edst[e]; }
  else             { src = e - N_EDGES; dst = src; }

  float4 l = xl[(size_t)src * 32 + lane];
  float4 r = xr[(size_t)dst * 32 + lane];
  float4 a = att4[lane];
  float v, t = 0.f;
  v = l.x + r.x; v = v > 0.f ? v : NEG_SLOPE * v; t += v * a.x;
  v = l.y + r.y; v = v > 0.f ? v : NEG_SLOPE * v; t += v * a.y;
  v = l.z + r.z; v = v > 0.f ? v : NEG_SLOPE * v; t += v * a.z;
  v = l.w + r.w; v = v > 0.f ? v : NEG_SLOPE * v; t += v * a.w;
  // reduce within 8-lane head group
  t += __shfl_xor(t, 1, 32);
  t += __shfl_xor(t, 2, 32);
  t += __shfl_xor(t, 4, 32);
  if ((lane & 7) == 0) {
    int h = lane >> 3;
    logit[(size_t)e * NH + h] = t;
    atomicMax(&mord[dst * NH + h], ordF(t));
  }
}

// ---------------- edge pass 2: a = exp(logit - m), s += a -----------
__global__ void k_edge_exp(float* __restrict__ logit, const int* __restrict__ edst,
                           const unsigned* __restrict__ mord, float* __restrict__ s) {
  int id = blockIdx.x * 256 + threadIdx.x;
  if (id >= ET * NH) return;
  int e = id >> 2, h = id & 3;
  int dst = (e < N_EDGES) ? edst[e] : (e - N_EDGES);
  float a = expf(logit[id] - unordF(mord[dst * NH + h]));
  logit[id] = a;                      // reuse buffer for numerators
  atomicAdd(&s[dst * NH + h], a);
}

// ---------------- edge pass 3: acc[dst] += alpha * xl[src] ----------
__global__ void k_edge_aggr(const float4* __restrict__ xl, const float* __restrict__ anum,
                            const int* __restrict__ esrc, const int* __restrict__ edst,
                            const float* __restrict__ s, float* __restrict__ acc) {
  const int wid = threadIdx.x >> 5, lane = threadIdx.x & 31;
  const int e = blockIdx.x * 8 + wid;
  if (e >= ET) return;
  int src, dst;
  if (e < N_EDGES) { src = esrc[e]; dst = edst[e]; }
  else             { src = e - N_EDGES; dst = src; }
  const int h = lane >> 3;
  float alpha = anum[(size_t)e * NH + h] / (s[dst * NH + h] + 1e-16f);
  float4 l = xl[(size_t)src * 32 + lane];
  float* base = acc + (size_t)dst * HC + lane * 4;
  atomicAdd(base + 0, alpha * l.x);
  atomicAdd(base + 1, alpha * l.y);
  atomicAdd(base + 2, alpha * l.z);
  atomicAdd(base + 3, alpha * l.w);
}

// ---------------- node-wise: +bias, batchnorm(eval), relu -----------
__global__ void k_bias_bn_relu(float* __restrict__ h, const float* __restrict__ bias,
                               const float* __restrict__ gamma, const float* __restrict__ beta,
                               const float* __restrict__ mean, const float* __restrict__ var) {
  int idx = blockIdx.x * 256 + threadIdx.x;
  if (idx >= N_NODES * HC) return;
  int ch = idx & (HC - 1);
  float v  = h[idx] + bias[ch];
  float sc = gamma[ch] * rsqrtf(var[ch] + BN_EPS);
  v = (v - mean[ch]) * sc + beta[ch];
  h[idx] = v > 0.f ? v : 0.f;
}

// ---------------- global mean pool (sums + counts via atomics) ------
__global__ void k_pool(const float4* __restrict__ h, const int* __restrict__ batch,
                       float* __restrict__ sums, float* __restrict__ cnts) {
  const int wid = threadIdx.x >> 5, lane = threadIdx.x & 31;
  const int n = blockIdx.x * 8 + wid;
  if (n >= N_NODES) return;
  int b = batch[n];
  float4 v = h[(size_t)n * 32 + lane];
  float* base = sums + (size_t)b * HC + lane * 4;
  atomicAdd(base + 0, v.x);
  atomicAdd(base + 1, v.y);
  atomicAdd(base + 2, v.z);
  atomicAdd(base + 3, v.w);
  if (lane == 0) atomicAdd(&cnts[b], 1.f);
}

// ---------------- fused MLP head: relu(comb @ W1 + b1) @ W2 + b2 ----
// one wave per batch element; thread c = hidden unit c (C=32)
__global__ void k_fc(const float* __restrict__ sums, const float* __restrict__ cnts,
                     const float* __restrict__ gf, const float* __restrict__ W1,
                     const float* __restrict__ b1, const float* __restrict__ W2,
                     const float* __restrict__ b2, float* __restrict__ out) {
  const int b = blockIdx.x;
  const int c = threadIdx.x;                    // 0..31
  float inv = 1.f / fmaxf(cnts[b], 1.f);
  float acc = b1[c];
  for (int k = 0; k < HC; ++k)
    acc += (sums[(size_t)b * HC + k] * inv) * W1[k * CC + c];
  for (int k = 0; k < G_IN; ++k)
    acc += gf[b * G_IN + k] * W1[(HC + k) * CC + c];
  acc = acc > 0.f ? acc : 0.f;
  float p = acc * W2[c];
  p += __shfl_xor(p, 1, 32);
  p += __shfl_xor(p, 2, 32);
  p += __shfl_xor(p, 4, 32);
  p += __shfl_xor(p, 8, 32);
  p += __shfl_xor(p, 16, 32);
  if (c == 0) out[b] = p + b2[0];
}

// ---------------- host side ----------------
extern "C" void kernel_launch(void* const* d_in, const int* in_sizes, int n_in,
                              void* d_out, int out_size, void* d_ws, size_t ws_size,
                              hipStream_t stream) {
  (void)in_sizes; (void)n_in; (void)out_size; (void)ws_size;
  // input order (setup_inputs insertion order, nested dicts flattened in order):
  // 0:x 1:global_feat, per layer i: [2+10i..] Wl,bl,Wr,br,att,bias, gamma,beta,mean,var
  // 32:fc1_W 33:fc1_b 34:fc2_W 35:fc2_b 36:edge_src 37:edge_dst 38:batch
  const float* x     = (const float*)d_in[0];
  const float* gf    = (const float*)d_in[1];
  const float* fc1W  = (const float*)d_in[32];
  const float* fc1b  = (const float*)d_in[33];
  const float* fc2W  = (const float*)d_in[34];
  const float* fc2b  = (const float*)d_in[35];
  const int*   esrc  = (const int*)d_in[36];
  const int*   edst  = (const int*)d_in[37];
  const int*   batch = (const int*)d_in[38];

  // workspace layout (floats)
  const size_t NF = (size_t)N_NODES * HC;
  float* ws   = (float*)d_ws;
  float* bufs[3] = { ws, ws + NF, ws + 2 * NF };       // rotating N x 128 buffers
  float*    logit = ws + 3 * NF;                        // ET*4
  unsigned* mord  = (unsigned*)(logit + (size_t)ET * NH); // N*4
  float*    svec  = (float*)(mord + (size_t)N_NODES * NH); // N*4
  float*    sums  = svec + (size_t)N_NODES * NH;        // B*128
  float*    cnts  = sums + (size_t)NB * HC;             // B

  // buffer rotation: layer0 xl=b0 xr/acc=b1 ; layer1 xl=b2 xr/acc=b0 ; layer2 xl=b1 xr/acc=b2
  const int xl_idx[3] = {0, 2, 1};
  const int xa_idx[3] = {1, 0, 2};

  const int gRows   = N_NODES / 16;                    // 3125 (exact)
  const int gEdgeW  = (ET + 7) / 8;                    // wave-per-edge kernels
  const int gEdgeT  = (ET * NH + 255) / 256;
  const int gNodeCh = (N_NODES * HC + 255) / 256;
  const int gNodeW  = (N_NODES + 7) / 8;

  const float* hin = x;
  for (int layer = 0; layer < 3; ++layer) {
    const float* Wl    = (const float*)d_in[2 + layer * 10 + 0];
    const float* bl    = (const float*)d_in[2 + layer * 10 + 1];
    const float* Wr    = (const float*)d_in[2 + layer * 10 + 2];
    const float* br    = (const float*)d_in[2 + layer * 10 + 3];
    const float* att   = (const float*)d_in[2 + layer * 10 + 4];
    const float* bias  = (const float*)d_in[2 + layer * 10 + 5];
    const float* gamma = (const float*)d_in[2 + layer * 10 + 6];
    const float* beta  = (const float*)d_in[2 + layer * 10 + 7];
    const float* mean  = (const float*)d_in[2 + layer * 10 + 8];
    const float* var   = (const float*)d_in[2 + layer * 10 + 9];
    float* xl = bufs[xl_idx[layer]];
    float* xa = bufs[xa_idx[layer]];   // holds xr, later reused as accumulator

    // linear transforms (fp32 WMMA)
    if (layer == 0) {
      k_gemm_wmma<F_IN><<<gRows, 256, 0, stream>>>(hin, Wl, bl, xl);
      k_gemm_wmma<F_IN><<<gRows, 256, 0, stream>>>(hin, Wr, br, xa);
    } else {
      k_gemm_wmma<HC><<<gRows, 256, 0, stream>>>(hin, Wl, bl, xl);
      k_gemm_wmma<HC><<<gRows, 256, 0, stream>>>(hin, Wr, br, xa);
    }

    // segment-max init (0 == minimum in ordered-u32 domain)
    k_fill<<<(N_NODES * NH + 255) / 256, 256, 0, stream>>>(mord, 0u, N_NODES * NH);
    // edge logits + per-target max
    k_edge_logit<<<gEdgeW, 256, 0, stream>>>((const float4*)xl, (const float4*)xa,
                                             esrc, edst, (const float4*)att, logit, mord);
    // softmax denominators
    k_fill<<<(N_NODES * NH + 255) / 256, 256, 0, stream>>>((unsigned*)svec, 0u, N_NODES * NH);
    k_edge_exp<<<gEdgeT, 256, 0, stream>>>(logit, edst, mord, svec);
    // xr is dead now -> reuse its buffer as the output accumulator
    k_fill<<<gNodeCh, 256, 0, stream>>>((unsigned*)xa, 0u, N_NODES * HC);
    k_edge_aggr<<<gEdgeW, 256, 0, stream>>>((const float4*)xl, logit, esrc, edst, svec, xa);
    // + bias, batchnorm(eval), relu  -> next layer input
    k_bias_bn_relu<<<gNodeCh, 256, 0, stream>>>(xa, bias, gamma, beta, mean, var);
    hin = xa;
  }

  // global mean pool + MLP head
  k_fill<<<(NB * HC + 255) / 256, 256, 0, stream>>>((unsigned*)sums, 0u, NB * HC);
  k_fill<<<(NB + 255) / 256, 256, 0, stream>>>((unsigned*)cnts, 0u, NB);
  k_pool<<<gNodeW, 256, 0, stream>>>((const float4*)hin, batch, sums, cnts);
  k_fc<<<NB, 32, 0, stream>>>(sums, cnts, gf, fc1W, fc1b, fc2W, fc2b, (float*)d_out);
}